// HybridGaussianFMeanLayer_18786186953010
// MI455X (gfx1250) — compile-verified
//
#include <hip/hip_runtime.h>
#include <hip/hip_bf16.h>
#include <math.h>

// CDNA5 / gfx1250. wave32.
typedef float v2f __attribute__((ext_vector_type(2)));
typedef float v4f __attribute__((ext_vector_type(4)));
typedef float v8f __attribute__((ext_vector_type(8)));

#define DDIM 1024
#define BDIM 32
#define EPSF 1e-8f

// ---------------------------------------------------------------------------
// Kernel 1: lin[b,o] = sum_k x[b,k] * W[o,k]   (fp32 WMMA, 16x16 tile / wave)
// Grid: 128 blocks x 32 threads (1 wave per 16x16 output tile, EXEC all ones).
// A (16x4 f32): lane L (m=L&15, half=L>>4): a = { A[m][2*half], A[m][2*half+1] }
// B (4x16 f32): lane L (n=L&15, half=L>>4): b = { B[2*half][n], B[2*half+1][n] }
// D (16x16 f32): VGPR r, lane L -> D[r + 8*half][L&15]
// ---------------------------------------------------------------------------
__global__ __launch_bounds__(32)
void hgfm_wmma_dot_kernel(const float* __restrict__ x,
                          const float* __restrict__ w,
                          float* __restrict__ lin)
{
    const int tile  = blockIdx.x;      // 0..127
    const int bTile = tile >> 6;       // 0..1   (B=32 -> two 16-row tiles)
    const int oTile = tile & 63;       // 0..63  (O=1024 -> 64 16-col tiles)
    const int lane  = threadIdx.x;     // 0..31
    const int half  = lane >> 4;       // 0 or 1
    const int l16   = lane & 15;

    // A row m = l16 comes from x; B column n = l16 comes from row (o0+n) of W.
    const float* __restrict__ xrow = x + (size_t)(bTile * 16 + l16) * DDIM;
    const float* __restrict__ wrow = w + (size_t)(oTile * 16 + l16) * DDIM;

    v8f c = {};
    #pragma unroll 4
    for (int k0 = 0; k0 < DDIM; k0 += 4) {
        const int ko = k0 + 2 * half;                 // 8-byte aligned
        v2f a = *(const v2f*)(xrow + ko);             // A[m][ko], A[m][ko+1]
        v2f b = *(const v2f*)(wrow + ko);             // B[ko][n], B[ko+1][n]
        // (neg_a, A, neg_b, B, c_mod, C, reuse_a, reuse_b)
        c = __builtin_amdgcn_wmma_f32_16x16x4_f32(false, a, false, b,
                                                  (short)0, c, false, false);
    }

    const int o = oTile * 16 + l16;
    #pragma unroll
    for (int r = 0; r < 8; ++r) {
        const int b = bTile * 16 + r + 8 * half;
        lin[(size_t)b * DDIM + o] = c[r];
    }
}

// ---------------------------------------------------------------------------
// Kernel 2: fmean path + final combine.
// One block (256 thr = 8 waves) per output column o:
//   - stage W[o,:] (4KB) into LDS once (256 x float4)
//   - wave w handles batch rows b = 4*w .. 4*w+3 (8 waves x 4 = all 32 rows)
//   - lane processes 4 consecutive i per iteration (float4 x-loads, LDS w)
//   z      = x[b,i] * W[o,i]
//   sp     = softplus(z) = max(z,0) + log1p(exp(-|z|))       (stable form)
//   t      = (sp + eps)^p[o]
//   fmean  = (sum_i t*z) / (sum_i t + eps)
//   gauss  = lin[b,o]          (softmax row-sums == 1 => gaussian path == dot)
//   out    = a0*(lin + bias[o]) + a1*fmean + a2*lin,  a = softmax(alphas[o,:])
// ---------------------------------------------------------------------------
__global__ __launch_bounds__(256)
void hgfm_fmean_combine_kernel(const float* __restrict__ x,
                               const float* __restrict__ w,
                               const float* __restrict__ bias,
                               const float* __restrict__ p,
                               const float* __restrict__ alphas,
                               const float* __restrict__ lin,
                               float* __restrict__ out)
{
    __shared__ float wsh[DDIM];                 // 4 KB: one W row
    const int o   = blockIdx.x;                 // 0..1023
    const int tid = threadIdx.x;

    // Cooperative stage of W[o,:] into LDS (b128 per thread).
    *(v4f*)(wsh + tid * 4) = *(const v4f*)(w + (size_t)o * DDIM + tid * 4);
    __syncthreads();

    const int wave = tid >> 5;                  // 0..7
    const int lane = tid & 31;
    const int b0   = wave * 4;

    const float po = p[o];
    const float* __restrict__ xr0 = x + (size_t)b0 * DDIM;

    float s1[4] = {0.f, 0.f, 0.f, 0.f};         // sum t      per batch row
    float s2[4] = {0.f, 0.f, 0.f, 0.f};         // sum t*z    per batch row

    #pragma unroll 2
    for (int it = 0; it < DDIM / 128; ++it) {   // 8 iterations
        const int i = it * 128 + lane * 4;
        const v4f wv = *(const v4f*)(wsh + i);  // ds_load_b128
        #pragma unroll
        for (int bb = 0; bb < 4; ++bb) {
            const v4f xv = *(const v4f*)(xr0 + (size_t)bb * DDIM + i);
            #pragma unroll
            for (int e = 0; e < 4; ++e) {
                const float z  = xv[e] * wv[e];
                const float sp = fmaxf(z, 0.0f) + log1pf(expf(-fabsf(z)));
                const float t  = powf(sp + EPSF, po);
                s1[bb] += t;
                s2[bb] = fmaf(t, z, s2[bb]);
            }
        }
    }

    // Mixing weights: uniform per block (o fixed), compute on every lane.
    const float a0 = alphas[o * 3 + 0];
    const float a1 = alphas[o * 3 + 1];
    const float a2 = alphas[o * 3 + 2];
    const float m  = fmaxf(a0, fmaxf(a1, a2));
    const float e0 = expf(a0 - m);
    const float e1 = expf(a1 - m);
    const float e2 = expf(a2 - m);
    const float inv = 1.0f / (e0 + e1 + e2);
    const float bo  = bias[o];

    #pragma unroll
    for (int bb = 0; bb < 4; ++bb) {
        float a = s1[bb], c = s2[bb];
        #pragma unroll
        for (int off = 16; off > 0; off >>= 1) {    // wave32 tree reduction
            a += __shfl_down(a, (unsigned)off, 32);
            c += __shfl_down(c, (unsigned)off, 32);
        }
        if (lane == 0) {
            const int   b  = b0 + bb;
            const float fm = c / (a + EPSF);
            const float d  = lin[(size_t)b * DDIM + o];   // WMMA dot
            out[(size_t)b * DDIM + o] =
                (e0 * (d + bo) + e1 * fm + e2 * d) * inv;
        }
    }
}

// ---------------------------------------------------------------------------
// Launch
// d_in order: x[B*D], weights[D*D], bias[D], p[D], log_sigma[D] (unused),
//             alphas[D*3]. d_out: [B*D] fp32. d_ws: lin scratch (B*D fp32).
// ---------------------------------------------------------------------------
extern "C" void kernel_launch(void* const* d_in, const int* in_sizes, int n_in,
                              void* d_out, int out_size, void* d_ws, size_t ws_size,
                              hipStream_t stream)
{
    const float* x      = (const float*)d_in[0];
    const float* wgt    = (const float*)d_in[1];
    const float* bias   = (const float*)d_in[2];
    const float* p      = (const float*)d_in[3];
    // d_in[4] = log_sigma: drops out analytically (softmax rows sum to 1)
    const float* alphas = (const float*)d_in[5];
    float* out = (float*)d_out;
    float* lin = (float*)d_ws;   // B*D floats = 128 KB

    // Kernel 1: 128 16x16 tiles, one wave each (fp32 WMMA dot).
    hipLaunchKernelGGL(hgfm_wmma_dot_kernel, dim3(128), dim3(32), 0, stream,
                       x, wgt, lin);

    // Kernel 2: one block per o; 8 waves x 4 batch rows each.
    hipLaunchKernelGGL(hgfm_fmean_combine_kernel, dim3(DDIM), dim3(256), 0, stream,
                       x, wgt, bias, p, alphas, lin, out);
}